// AuxCrossAttention_30537217474899
// MI455X (gfx1250) — compile-verified
//
#include <hip/hip_runtime.h>

typedef __attribute__((ext_vector_type(2))) float v2f;
typedef __attribute__((ext_vector_type(8))) float v8f;

#define B_    8
#define T_    128
#define C_    256
#define E2_   32
#define H_    8
#define HD_   32
#define WKLD  320           // C + 2*E2, row stride of Wk/Wv
#define SCALE 0.17677669529663687f  // 1/sqrt(32)

__device__ __forceinline__ v8f wmma4(v2f a, v2f b, v8f c) {
  // V_WMMA_F32_16X16X4_F32: D = A(16x4) * B(4x16) + C(16x16), fp32 throughout
  return __builtin_amdgcn_wmma_f32_16x16x4_f32(false, a, false, b, (short)0, c,
                                               false, false);
}

// acc += A[mBase:mBase+16, 0:K] * W[nBase:nBase+16, 0:K]^T
// A[m][k] = A[m*lda + k]; W[n][k] = W[n*ldw + k] (i.e. B-matrix = W^T)
// Fragment layout (ISA 7.12.2, 32-bit): A lane(l16,half) holds K = 2*half + r,
// B same, C/D row = r + 8*half, col = l16.
__device__ __forceinline__ void gemm_tile(const float* __restrict__ A, int lda,
                                          const float* __restrict__ W, int ldw,
                                          int K, int mBase, int nBase,
                                          int l16, int half, v8f& acc) {
  const float* ap = A + (mBase + l16) * lda + 2 * half;
  const float* wp = W + (nBase + l16) * ldw + 2 * half;
#pragma unroll 4
  for (int k0 = 0; k0 < K; k0 += 4) {
    v2f a = *(const v2f*)(ap + k0);
    v2f b = *(const v2f*)(wp + k0);
    acc = wmma4(a, b, acc);
  }
}

// ---------------- Phase 1: projections q, ks, kt, vs, vt ----------------
__global__ __launch_bounds__(32) void proj_kernel(
    const float* __restrict__ x1, const float* __restrict__ x2,
    const float* __restrict__ aux1, const float* __restrict__ aux2,
    const float* __restrict__ Wq, const float* __restrict__ bq,
    const float* __restrict__ Wk, const float* __restrict__ bk,
    const float* __restrict__ Wv, const float* __restrict__ bv,
    float* __restrict__ q, float* __restrict__ ks, float* __restrict__ kt,
    float* __restrict__ vs, float* __restrict__ vt) {
  const int lane = threadIdx.x & 31;
  const int half = lane >> 4, l16 = lane & 15;
  const int nBase = blockIdx.x * 16;
  const int mBase = blockIdx.y * 16;
  const int b = blockIdx.z / 5;
  const int which = blockIdx.z % 5;

  v8f acc = {0.f, 0.f, 0.f, 0.f, 0.f, 0.f, 0.f, 0.f};
  const float* bias = nullptr;
  float* outp = nullptr;

  if (which == 0) {            // q = x1 @ Wq^T + bq
    gemm_tile(x1 + b * T_ * C_, C_, Wq, C_, C_, mBase, nBase, l16, half, acc);
    bias = bq; outp = q;
  } else if (which == 1) {     // ks = x2 @ Wk[:,0:256]^T + aux2 @ Wk[:,288:]^T + bk
    gemm_tile(x2 + b * T_ * C_, C_, Wk, WKLD, C_, mBase, nBase, l16, half, acc);
    gemm_tile(aux2 + b * T_ * E2_, E2_, Wk + C_ + E2_, WKLD, E2_, mBase, nBase, l16, half, acc);
    bias = bk; outp = ks;
  } else if (which == 2) {     // kt = aux1 @ Wk[:,256:288]^T
    gemm_tile(aux1 + b * T_ * E2_, E2_, Wk + C_, WKLD, E2_, mBase, nBase, l16, half, acc);
    outp = kt;
  } else if (which == 3) {     // vs
    gemm_tile(x2 + b * T_ * C_, C_, Wv, WKLD, C_, mBase, nBase, l16, half, acc);
    gemm_tile(aux2 + b * T_ * E2_, E2_, Wv + C_ + E2_, WKLD, E2_, mBase, nBase, l16, half, acc);
    bias = bv; outp = vs;
  } else {                     // vt
    gemm_tile(aux1 + b * T_ * E2_, E2_, Wv + C_, WKLD, E2_, mBase, nBase, l16, half, acc);
    outp = vt;
  }

  const float bval = bias ? bias[nBase + l16] : 0.0f;
  float* op = outp + b * T_ * C_;
#pragma unroll
  for (int r = 0; r < 8; ++r)
    op[(mBase + r + 8 * half) * C_ + nBase + l16] = acc[r] + bval;
}

// ---------------- Phase 2: attention per (b, h) ----------------
// Block = 8 waves; wave w owns rows [16w, 16w+16) of the 128x128 score matrix.
#define LDS_STRIDE 132  // 128 + 4 floats of padding (528B, 8B aligned rows)
__global__ __launch_bounds__(256) void attn_kernel(
    const float* __restrict__ q, const float* __restrict__ ks,
    const float* __restrict__ kt, const float* __restrict__ vs,
    const float* __restrict__ vt, const int* __restrict__ mask,
    float* __restrict__ y) {
  __shared__ __align__(16) float ldsP[T_ * LDS_STRIDE];  // softmax probs (transpose stage)

  const int h = blockIdx.x, b = blockIdx.y;
  const int w = threadIdx.x >> 5;
  const int lane = threadIdx.x & 31;
  const int half = lane >> 4, l16 = lane & 15;
  const int mBase = w * 16;
  const float NEG_INF = -__builtin_inff();

  const float* qh  = q  + b * T_ * C_ + h * HD_;  // qh[m][d]  = qh[m*C_+d]
  const float* ksh = ks + b * T_ * C_ + h * HD_;  // ks[s][d]
  const float* vsh = vs + b * T_ * C_ + h * HD_;  // vs[s][d]
  const float* kth = kt + b * T_ * C_ + h * HD_;
  const float* vth = vt + b * T_ * C_ + h * HD_;

  // Per-row constant (q . kt) * scale — invariant over s; lanes 0..15 and
  // 16..31 redundantly compute row mBase+l16, distributed below via shuffle.
  float rowdot = 0.f;
  {
    const float* qr = qh + (mBase + l16) * C_;
    const float* kr = kth + (mBase + l16) * C_;
#pragma unroll
    for (int d = 0; d < HD_; ++d) rowdot += qr[d] * kr[d];
    rowdot *= SCALE;
  }
  float rc[8];
#pragma unroll
  for (int r = 0; r < 8; ++r) rc[r] = __shfl(rowdot, 8 * half + r, 32);

  // Scores: 8 tiles of 16x16 along s, K = 32 (head dim)
  v8f s[8];
#pragma unroll
  for (int j = 0; j < 8; ++j) {
    s[j] = (v8f){0.f, 0.f, 0.f, 0.f, 0.f, 0.f, 0.f, 0.f};
    gemm_tile(qh, C_, ksh, C_, HD_, mBase, j * 16, l16, half, s[j]);
  }

  // Scale + row const + mask
  const int* mrow = mask + b * T_ * T_;
#pragma unroll
  for (int j = 0; j < 8; ++j) {
#pragma unroll
    for (int r = 0; r < 8; ++r) {
      const int row = mBase + r + 8 * half;
      const int col = j * 16 + l16;
      float v = s[j][r] * SCALE + rc[r];
      if (mrow[row * T_ + col] == 0) v = NEG_INF;
      s[j][r] = v;
    }
  }

  // Row-wise softmax: reduce over 8 local tiles + 16 lanes of the half-wave.
#pragma unroll
  for (int r = 0; r < 8; ++r) {
    float mx = s[0][r];
#pragma unroll
    for (int j = 1; j < 8; ++j) mx = fmaxf(mx, s[j][r]);
#pragma unroll
    for (int o = 1; o < 16; o <<= 1) mx = fmaxf(mx, __shfl_xor(mx, o, 32));
    const bool allm = (mx == NEG_INF);  // fully-masked row -> uniform (ref behavior)
    const float sub = allm ? 0.f : mx;
    float sum = 0.f;
#pragma unroll
    for (int j = 0; j < 8; ++j) {
      float lv = allm ? 0.f : s[j][r];
      float p = __expf(lv - sub);
      s[j][r] = p;
      sum += p;
    }
#pragma unroll
    for (int o = 1; o < 16; o <<= 1) sum += __shfl_xor(sum, o, 32);
    const float inv = 1.0f / sum;
    const int row = mBase + r + 8 * half;
#pragma unroll
    for (int j = 0; j < 8; ++j)
      ldsP[row * LDS_STRIDE + j * 16 + l16] = s[j][r] * inv;  // C-layout -> LDS
  }
  __syncthreads();

  // Y = P @ vs_h  (16 x 32 output, K = 128), re-reading P in A-fragment layout
  const float* ap = &ldsP[(mBase + l16) * LDS_STRIDE + 2 * half];
#pragma unroll
  for (int jn = 0; jn < 2; ++jn) {
    v8f acc = {0.f, 0.f, 0.f, 0.f, 0.f, 0.f, 0.f, 0.f};
    const int n = jn * 16 + l16;
#pragma unroll 4
    for (int k0 = 0; k0 < T_; k0 += 4) {
      v2f a = *(const v2f*)(ap + k0);                 // ds_load_b64
      const int kb = k0 + 2 * half;
      v2f bb;
      bb.x = vsh[kb * C_ + n];
      bb.y = vsh[(kb + 1) * C_ + n];
      acc = wmma4(a, bb, acc);
    }
#pragma unroll
    for (int r = 0; r < 8; ++r) {
      const int row = mBase + r + 8 * half;
      y[b * T_ * C_ + row * C_ + h * HD_ + n] = acc[r] + vth[row * C_ + n];
    }
  }
}

// ---------------- Phase 3: out = y @ Wc^T + bc ----------------
__global__ __launch_bounds__(32) void outproj_kernel(
    const float* __restrict__ y, const float* __restrict__ Wc,
    const float* __restrict__ bc, float* __restrict__ out) {
  const int lane = threadIdx.x & 31;
  const int half = lane >> 4, l16 = lane & 15;
  const int nBase = blockIdx.x * 16;
  const int mBase = blockIdx.y * 16;
  const int b = blockIdx.z;

  v8f acc = {0.f, 0.f, 0.f, 0.f, 0.f, 0.f, 0.f, 0.f};
  gemm_tile(y + b * T_ * C_, C_, Wc, C_, C_, mBase, nBase, l16, half, acc);

  const float bval = bc[nBase + l16];
  float* op = out + b * T_ * C_;
#pragma unroll
  for (int r = 0; r < 8; ++r)
    op[(mBase + r + 8 * half) * C_ + nBase + l16] = acc[r] + bval;
}

extern "C" void kernel_launch(void* const* d_in, const int* in_sizes, int n_in,
                              void* d_out, int out_size, void* d_ws, size_t ws_size,
                              hipStream_t stream) {
  (void)in_sizes; (void)n_in; (void)out_size; (void)ws_size;
  const float* x1   = (const float*)d_in[0];
  const float* x2   = (const float*)d_in[1];
  const int*   mask = (const int*)d_in[2];
  const float* aux1 = (const float*)d_in[3];
  const float* aux2 = (const float*)d_in[4];
  const float* Wq   = (const float*)d_in[5];
  const float* bq   = (const float*)d_in[6];
  const float* Wk   = (const float*)d_in[7];
  const float* bk   = (const float*)d_in[8];
  const float* Wv   = (const float*)d_in[9];
  const float* bv   = (const float*)d_in[10];
  const float* Wc   = (const float*)d_in[11];
  const float* bc   = (const float*)d_in[12];
  float* out = (float*)d_out;

  const size_t MAT = (size_t)B_ * T_ * C_;  // 262144 floats
  float* q  = (float*)d_ws;
  float* ks = q  + MAT;
  float* kt = ks + MAT;
  float* vs = kt + MAT;
  float* vt = vs + MAT;
  float* y  = vt + MAT;

  // Phase 1: 16 n-tiles x 8 m-tiles x (B * 5 outputs), one wave each
  proj_kernel<<<dim3(C_ / 16, T_ / 16, B_ * 5), 32, 0, stream>>>(
      x1, x2, aux1, aux2, Wq, bq, Wk, bk, Wv, bv, q, ks, kt, vs, vt);

  // Phase 2: one block per (h, b), 8 waves
  attn_kernel<<<dim3(H_, B_), 256, 0, stream>>>(q, ks, kt, vs, vt, mask, y);

  // Phase 3: output projection
  outproj_kernel<<<dim3(C_ / 16, T_ / 16, B_), 32, 0, stream>>>(y, Wc, bc, out);
}